// LightGCN_37924561223760
// MI455X (gfx1250) — compile-verified
//
#include <hip/hip_runtime.h>
#include <math.h>

#define N_USERS 100000
#define N_ITEMS 50000
#define N_NODES (N_USERS + N_ITEMS)
#define DIM     64
#define N_HOPS  3
#define NNZ     3200000
#define BATCH   4096
#define KNEG    4
#define DECAYF  1e-4f

typedef __attribute__((ext_vector_type(2))) float v2f;
typedef __attribute__((ext_vector_type(8))) float v8f;

__device__ __forceinline__ void atomAddF(float* p, float v) {
  // hardware global_atomic_add_f32 path
  unsafeAtomicAdd(p, v);
}

__global__ void k_init_out(float* __restrict__ out) {
  if (blockIdx.x == 0 && threadIdx.x == 0) out[0] = 0.0f;
}

// Build e0 = concat(user_embed, item_embed) as [N_NODES, 64] f32, float4 lanes.
__global__ void k_concat(const float* __restrict__ ue, const float* __restrict__ ie,
                         float* __restrict__ dst) {
  unsigned t = blockIdx.x * blockDim.x + threadIdx.x;
  if (t >= (unsigned)N_NODES * 16u) return;
  unsigned row = t >> 4, q = t & 15u;
  const float* s = (row < (unsigned)N_USERS) ? (ue + (size_t)row * DIM)
                                             : (ie + (size_t)(row - N_USERS) * DIM);
  ((float4*)dst)[t] = *(const float4*)(s + q * 4);
}

__global__ void k_zero_nodes(float* __restrict__ dst) {
  unsigned t = blockIdx.x * blockDim.x + threadIdx.x;
  if (t >= (unsigned)N_NODES * 16u) return;
  ((float4*)dst)[t] = make_float4(0.f, 0.f, 0.f, 0.f);
}

// Scatter SpMM: next[rows[e]] += vals[e] * cur[cols[e]], 16 lanes per edge, float4/lane.
// Both buffers (77 MB) are L2-resident; atomics resolve at L2.
__global__ void k_spmm(const float* __restrict__ cur, float* __restrict__ nxt,
                       const float* __restrict__ vals, const int* __restrict__ rows,
                       const int* __restrict__ cols) {
  unsigned t = blockIdx.x * blockDim.x + threadIdx.x;
  unsigned e = t >> 4;
  if (e >= (unsigned)NNZ) return;
  unsigned q = t & 15u;
  // light prefetch of upcoming edge metadata (global_prefetch_b8)
  if (q == 0 && e + 2048u < (unsigned)NNZ) {
    __builtin_prefetch(cols + e + 2048, 0, 0);
    __builtin_prefetch(rows + e + 2048, 0, 0);
  }
  int   r = rows[e];
  int   c = cols[e];
  float v = vals[e];
  float4 s = *(const float4*)(cur + (size_t)c * DIM + q * 4);
  float* d = nxt + (size_t)r * DIM + q * 4;
  atomAddF(d + 0, v * s.x);
  atomAddF(d + 1, v * s.y);
  atomAddF(d + 2, v * s.z);
  atomAddF(d + 3, v * s.w);
}

// Gather batch rows from src ([N_NODES,64] concat layout) into per-batch hop-sum
// accumulators. first=1: overwrite and also accumulate the L2 regularizer term
// (reg only uses hop-0 rows). first=0: add.
__global__ void k_gather(const float* __restrict__ src,
                         const int* __restrict__ users, const int* __restrict__ pos,
                         const int* __restrict__ neg,
                         float* __restrict__ u_acc, float* __restrict__ p_acc,
                         float* __restrict__ n_acc,
                         float* __restrict__ d_out, int first) {
  __shared__ float red[256];
  unsigned t = blockIdx.x * blockDim.x + threadIdx.x;
  const unsigned TOT = (unsigned)(BATCH * (2 + KNEG)) * 16u;  // 24576 rows * 16
  float sq = 0.f;
  if (t < TOT) {
    unsigned idx = t >> 4, q = t & 15u;
    size_t srow;
    float* dst;
    if (idx < (unsigned)BATCH) {
      srow = (size_t)users[idx];
      dst  = u_acc + (size_t)idx * DIM;
    } else if (idx < 2u * BATCH) {
      unsigned j = idx - BATCH;
      srow = (size_t)N_USERS + (size_t)pos[j];
      dst  = p_acc + (size_t)j * DIM;
    } else {
      unsigned j = idx - 2u * BATCH;           // j in [0, BATCH*KNEG)
      srow = (size_t)N_USERS + (size_t)neg[j];
      dst  = n_acc + (size_t)j * DIM;
    }
    float4 v = *(const float4*)(src + srow * DIM + q * 4);
    float4* dp = (float4*)(dst + q * 4);
    if (first) {
      *dp = v;
      sq = v.x * v.x + v.y * v.y + v.z * v.z + v.w * v.w;
    } else {
      float4 o = *dp;
      o.x += v.x; o.y += v.y; o.z += v.z; o.w += v.w;
      *dp = o;
    }
  }
  if (first) {
    red[threadIdx.x] = sq;
    __syncthreads();
    for (int s = 128; s > 0; s >>= 1) {
      if ((int)threadIdx.x < s) red[threadIdx.x] += red[threadIdx.x + s];
      __syncthreads();
    }
    if (threadIdx.x == 0)
      atomAddF(d_out, red[0] * (DECAYF / (2.0f * (float)BATCH)));
  }
}

__device__ __forceinline__ float softplusf(float x) {
  return fmaxf(x, 0.f) + log1pf(expf(-fabsf(x)));
}

// One wave32 per 16-row batch tile. Scores via chained V_WMMA_F32_16X16X4_F32:
// C = U_tile (16x64) x Item_tile^T (64x16); diagonal = the 16 dot products.
// Accumulators hold hop-SUMS; mean scaling (1/4 per operand) folded as 1/16.
__global__ void k_score(const float* __restrict__ u_acc,
                        const float* __restrict__ p_acc,
                        const float* __restrict__ n_acc,
                        float* __restrict__ d_out) {
  const int lane = threadIdx.x;         // 0..31, EXEC all ones for WMMA region
  const int tile = blockIdx.x;          // 0..BATCH/16-1
  const int idx  = lane & 15;           // A-row m / B-col n for this lane
  const int koff = (lane >> 4) * 2;     // K sub-offset per ISA A/B f32 layout
  const int rowB = tile * 16 + idx;     // batch row this lane loads

  const float* abase = u_acc + (size_t)rowB * DIM;

  // diag ownership: lanes 0..7 own m=lane (c[lane]); lanes 24..31 own m=lane-16 (c[lane-24])
  const bool own = (lane < 8) || (lane >= 24);
  const int  ci  = (lane < 8) ? lane : (lane - 24);

  float loss = 0.f;

#pragma unroll
  for (int item = 0; item < 1 + KNEG; ++item) {
    const float* bbase = (item == 0)
        ? (p_acc + (size_t)rowB * DIM)
        : (n_acc + ((size_t)rowB * KNEG + (item - 1)) * DIM);
    v8f c = {};
#pragma unroll
    for (int kk = 0; kk < 16; ++kk) {
      v2f a = *(const v2f*)(abase + 4 * kk + koff);
      v2f b = *(const v2f*)(bbase + 4 * kk + koff);
      c = __builtin_amdgcn_wmma_f32_16x16x4_f32(
          /*neg_a=*/false, a, /*neg_b=*/false, b,
          /*c_mod=*/(short)0, c, /*reuse_a=*/false, /*reuse_b=*/false);
    }
    if (own) {
      float s = 0.f;
#pragma unroll
      for (int i = 0; i < 8; ++i)
        if (ci == i) s = c[i];
      s *= 0.0625f;                         // (1/4 hop-mean) on each operand
      float x = (item == 0) ? -s : s;       // BCE-with-logits pos / neg terms
      loss += softplusf(x);
    }
  }

  // wave32 reduction of per-row losses (non-owning lanes contribute 0)
  for (int off = 16; off > 0; off >>= 1)
    loss += __shfl_down(loss, off, 32);
  if (lane == 0)
    atomAddF(d_out, loss * (1.0f / (float)BATCH));
}

extern "C" void kernel_launch(void* const* d_in, const int* in_sizes, int n_in,
                              void* d_out, int out_size, void* d_ws, size_t ws_size,
                              hipStream_t stream) {
  (void)in_sizes; (void)n_in; (void)out_size; (void)ws_size;
  const float* user_embed = (const float*)d_in[0];
  const float* item_embed = (const float*)d_in[1];
  const float* adj_vals   = (const float*)d_in[2];
  const int*   adj_rows   = (const int*)d_in[3];
  const int*   adj_cols   = (const int*)d_in[4];
  const int*   users      = (const int*)d_in[5];
  const int*   pos_items  = (const int*)d_in[6];
  const int*   neg_items  = (const int*)d_in[7];
  float* out = (float*)d_out;

  char* ws = (char*)d_ws;
  size_t off = 0;
  auto carve = [&](size_t bytes) -> float* {
    float* p = (float*)(ws + off);
    off += (bytes + 255) & ~(size_t)255;
    return p;
  };
  const size_t NODE_BYTES = (size_t)N_NODES * DIM * sizeof(float);   // 38.4 MB
  float* bufA  = carve(NODE_BYTES);
  float* bufB  = carve(NODE_BYTES);
  float* u_acc = carve((size_t)BATCH * DIM * sizeof(float));
  float* p_acc = carve((size_t)BATCH * DIM * sizeof(float));
  float* n_acc = carve((size_t)BATCH * KNEG * DIM * sizeof(float));

  const int TB = 256;
  const unsigned nodeGrid   = ((unsigned)N_NODES * 16u + TB - 1) / TB;   // 9375
  const unsigned spmmGrid   = ((unsigned)NNZ * 16u + TB - 1) / TB;       // 200000
  const unsigned gatherGrid = ((unsigned)(BATCH * (2 + KNEG)) * 16u + TB - 1) / TB; // 1536

  k_init_out<<<1, 64, 0, stream>>>(out);
  k_concat<<<nodeGrid, TB, 0, stream>>>(user_embed, item_embed, bufA);
  // hop 0 gather (overwrite) + regularizer on hop-0 rows
  k_gather<<<gatherGrid, TB, 0, stream>>>(bufA, users, pos_items, neg_items,
                                          u_acc, p_acc, n_acc, out, 1);

  float* cur = bufA;
  float* nxt = bufB;
  for (int h = 0; h < N_HOPS; ++h) {
    k_zero_nodes<<<nodeGrid, TB, 0, stream>>>(nxt);
    k_spmm<<<spmmGrid, TB, 0, stream>>>(cur, nxt, adj_vals, adj_rows, adj_cols);
    k_gather<<<gatherGrid, TB, 0, stream>>>(nxt, users, pos_items, neg_items,
                                            u_acc, p_acc, n_acc, out, 0);
    float* t = cur; cur = nxt; nxt = t;
  }

  k_score<<<BATCH / 16, 32, 0, stream>>>(u_acc, p_acc, n_acc, out);
}